// RoIAlign_31464930410632
// MI455X (gfx1250) — compile-verified
//
#include <hip/hip_runtime.h>

typedef __attribute__((ext_vector_type(2))) float v2f;
typedef __attribute__((ext_vector_type(8))) float v8f;

#define PHB 7
#define PWB 7
#define NS  14      // sample positions per dim
#define NSL 28      // slots per dim (2 corners per sample)
#define CT  16      // channel tile (WMMA M)
#define HH  200
#define WW  200
#define CC  256
#define PAD 30      // padded innermost stride (even -> 8B aligned b64 loads)

#define USE_ASYNC_GATHER 1

__device__ __forceinline__ void wait_async0() {
#if __has_builtin(__builtin_amdgcn_s_wait_asynccnt)
    __builtin_amdgcn_s_wait_asynccnt(0);
#else
    asm volatile("s_wait_asynccnt 0x0" ::: "memory");
#endif
}

__global__ __launch_bounds__(256, 2)
void roi_align_wmma_kernel(const float* __restrict__ feat,
                           const float* __restrict__ rois,
                           float* __restrict__ out)
{
    __shared__ __attribute__((aligned(16))) float Gs[2][NSL * CT * PAD]; // [buf][ys][ch][xs]
    __shared__ __attribute__((aligned(16))) float Txs[CT * PWB * PAD];   // [ch][px][ys]
    __shared__ __attribute__((aligned(8)))  float wx_s[NSL];
    __shared__ __attribute__((aligned(8)))  float wy_s[NSL];
    __shared__ int xcol_s[NSL];
    __shared__ int yrow_s[NSL];
    __shared__ int bidx_s;

    const int roi   = blockIdx.x;
    const int tid   = threadIdx.x;
    const int lane  = tid & 31;
    const int wave  = tid >> 5;
    const int halfw = lane >> 4;   // 0: K{0,1}, 1: K{2,3} (32-bit A/B layout)
    const int nn    = lane & 15;   // M index for A, N index for B/D

    // ---- per-RoI geometry (folded weights: 0.5 pooling factor + validity) ----
    const float x1f = rois[roi*5+1] * 0.0625f;
    const float y1f = rois[roi*5+2] * 0.0625f;
    const float x2f = rois[roi*5+3] * 0.0625f;
    const float y2f = rois[roi*5+4] * 0.0625f;
    const float roi_w = fmaxf(x2f - x1f, 1.0f);
    const float roi_h = fmaxf(y2f - y1f, 1.0f);
    const float bin_w = roi_w * (1.0f / PWB);
    const float bin_h = roi_h * (1.0f / PHB);

    if (tid == 0) bidx_s = (int)rois[roi*5+0];
    if (tid < 2*NS) {
        const bool  isY = tid < NS;
        const int   s   = isY ? tid : tid - NS;
        const float org = isY ? y1f : x1f;
        const float bsz = isY ? bin_h : bin_w;
        const float pos = org + (0.25f + 0.5f * (float)s) * bsz;   // (p + (j+0.5)/2)*bin
        const bool  valid = (pos >= -1.0f) && (pos <= 200.0f);
        float p  = fminf(fmaxf(pos, 0.0f), 199.0f);
        int   i0 = (int)floorf(p);
        int   i1 = (i0 + 1 < 199) ? (i0 + 1) : 199;
        float lf = p - (float)i0;
        float w0 = valid ? (1.0f - lf) * 0.5f : 0.0f;
        float w1 = valid ? lf * 0.5f : 0.0f;
        if (isY) { yrow_s[2*s] = i0; yrow_s[2*s+1] = i1; wy_s[2*s] = w0; wy_s[2*s+1] = w1; }
        else     { xcol_s[2*s] = i0; xcol_s[2*s+1] = i1; wx_s[2*s] = w0; wx_s[2*s+1] = w1; }
    }
    __syncthreads();

    const float* fbase = feat + (size_t)bidx_s * CC * (HH*WW);

    // ---- gather one 28x16x28 channel tile: async global -> LDS (ASYNCcnt path) ----
    auto issue_gather = [&](int tile, int buf) {
        const float* fb = fbase + (size_t)(tile * CT) * (HH*WW);
        const unsigned lds_base = (unsigned)(uintptr_t)&Gs[buf][0]; // low 32 bits = LDS offset
        #pragma unroll 7
        for (int it = 0; it < 49; ++it) {
            int e  = tid + it * 256;
            int xs = e % NSL;
            int r  = e / NSL;
            int ch = r & (CT - 1);
            int ys = r >> 4;
            unsigned ldsa = lds_base + (unsigned)(((ys*CT + ch)*PAD + xs) * 4);
            int      elt  = ch*(HH*WW) + yrow_s[ys]*WW + xcol_s[xs];
#if USE_ASYNC_GATHER
            const float* gp = fb + elt;   // per-lane 64-bit address (GV mode)
            asm volatile("global_load_async_to_lds_b32 %0, %1, off"
                         :: "v"(ldsa), "v"(gp) : "memory");
#else
            Gs[buf][(ys*CT + ch)*PAD + xs] = fb[elt];
#endif
        }
    };

    issue_gather(0, 0);

    for (int tile = 0; tile < CC / CT; ++tile) {
        const int cur = tile & 1;
        wait_async0();           // my wave's async writes for `tile` are in LDS
        __syncthreads();         // everyone's are; Txs consumers of tile-1 done

        if (tile + 1 < CC / CT) issue_gather(tile + 1, cur ^ 1); // overlap with WMMA

        // ---- matmul 1: x-interp  Tx(16ch x 7px) = G(16 x 28) * Wx(28 x 7), K=28 via 7x K4 WMMA ----
        for (int ys = wave; ys < NSL; ys += 8) {
            v8f acc = {};
            #pragma unroll
            for (int kk = 0; kk < 7; ++kk) {
                int k0 = 4*kk + 2*halfw;
                v2f a  = *(const v2f*)&Gs[cur][(ys*CT + nn)*PAD + k0];
                v2f wp = *(const v2f*)&wx_s[k0];
                v2f b;
                b.x = (kk == nn) ? wp.x : 0.0f;   // Wx[xslot, px]: nonzero iff xslot/4 == px
                b.y = (kk == nn) ? wp.y : 0.0f;
                acc = __builtin_amdgcn_wmma_f32_16x16x4_f32(
                        false, a, false, b, (short)0, acc, false, false);
            }
            if (nn < PWB) {
                #pragma unroll
                for (int v = 0; v < 8; ++v) {
                    // D layout: M = v + 8*halfw (channel), N = nn (px)
                    Txs[((v + 8*halfw)*PWB + nn)*PAD + ys] = acc[v];
                }
            }
        }
        __syncthreads();

        // ---- matmul 2: y-interp  out(16ch x 7py) = Tx(16 x 28) * Wy(28 x 7) per px ----
        if (wave < PWB) {
            const int px = wave;
            v8f acc = {};
            #pragma unroll
            for (int kk = 0; kk < 7; ++kk) {
                int k0 = 4*kk + 2*halfw;
                v2f a  = *(const v2f*)&Txs[(nn*PWB + px)*PAD + k0];
                v2f wp = *(const v2f*)&wy_s[k0];
                v2f b;
                b.x = (kk == nn) ? wp.x : 0.0f;   // Wy[yslot, py]: nonzero iff yslot/4 == py
                b.y = (kk == nn) ? wp.y : 0.0f;
                acc = __builtin_amdgcn_wmma_f32_16x16x4_f32(
                        false, a, false, b, (short)0, acc, false, false);
            }
            if (nn < PHB) {
                #pragma unroll
                for (int v = 0; v < 8; ++v) {
                    int ch = tile*CT + v + 8*halfw;
                    out[(((size_t)roi*CC + ch)*PHB + nn)*PWB + px] = acc[v];
                }
            }
        }
        // next iteration's wait+barrier orders Txs reuse and buffer swap
    }
}

extern "C" void kernel_launch(void* const* d_in, const int* in_sizes, int n_in,
                              void* d_out, int out_size, void* d_ws, size_t ws_size,
                              hipStream_t stream) {
    const float* feat = (const float*)d_in[0];
    const float* rois = (const float*)d_in[1];
    float* out = (float*)d_out;
    const int K = in_sizes[1] / 5;   // rois are (K,5)
    if (K <= 0) return;
    roi_align_wmma_kernel<<<dim3(K), dim3(256), 0, stream>>>(feat, rois, out);
}